// MSSSIM_50070728737622
// MI455X (gfx1250) — compile-verified
//
#include <hip/hip_runtime.h>
#include <math.h>

typedef float v2f __attribute__((ext_vector_type(2)));
typedef float v4f __attribute__((ext_vector_type(4)));
typedef float v8f __attribute__((ext_vector_type(8)));

#define NB 32   // batch
#define NC 3    // channels

// normalized gaussian tap: softmax(-(i-5)^2 / (2*1.5^2)); invs = 1/sum precomputed
__device__ __forceinline__ float wval(int i, float invs) {
    if ((unsigned)i >= 11u) return 0.0f;
    float d = (float)i - 5.0f;
    return __expf(-d * d / 4.5f) * invs;
}

#define WMMA_F32(A, B, C) \
    __builtin_amdgcn_wmma_f32_16x16x4_f32(false, (A), false, (B), (short)0, (C), false, false)

// ---------------------------------------------------------------------------
// SSIM tile kernel: one wave (32 threads) per 16x16 output tile of one image.
// Both separable conv passes run on V_WMMA_F32_16X16X4_F32 with a banded
// Gaussian matrix (K = 26 padded to 28; pad rows carry zero weights).
// 5 quantity chains (x, y, x^2, y^2, xy) are interleaved so consecutive
// WMMAs are independent (no D->C RAW stalls) and LDS traffic is amortized.
// ---------------------------------------------------------------------------
__global__ __launch_bounds__(32)
void ssim_level_kernel(const float* __restrict__ X, const float* __restrict__ Y,
                       int H, int W, int level,
                       const float* __restrict__ c1c2,
                       float* __restrict__ ssim_acc,   // [5][NB]
                       float* __restrict__ cs_acc)     // [5][NB]
{
    __shared__ float xs[32][28];        // input tile, zero-padded
    __shared__ float ys[32][28];
    __shared__ float hsT[5][16][36];    // hsum transposed: [q][col][row], stride 36

    const int lane = threadIdx.x;             // 0..31
    const int mh = H - 10, mw = W - 10;
    const int img = blockIdx.z;               // n*NC + c
    const int n = img / NC;
    const long imgoff = (long)img * H * W;
    const int ox0 = blockIdx.x * 16, oy0 = blockIdx.y * 16;

    // gaussian normalization
    float s = 0.f;
    #pragma unroll
    for (int t = 0; t < 11; ++t) { float d = (float)t - 5.f; s += __expf(-d * d / 4.5f); }
    const float invs = 1.f / s;

    // load 26x26 input tile; everything else (rows 26-31, cols 26-27, OOB) -> 0
    for (int idx = lane; idx < 32 * 28; idx += 32) {
        int r = idx / 28, c = idx % 28;
        int gr = oy0 + r, gc = ox0 + c;
        float xv = 0.f, yv = 0.f;
        if (r < 26 && c < 26 && gr < H && gc < W) {
            long o = imgoff + (long)gr * W + gc;
            xv = X[o];
            yv = Y[o];
        }
        xs[r][c] = xv;
        ys[r][c] = yv;
    }

    const int nlo   = lane & 15;          // N-col (B/C/D) or M-row (A) per ISA layout
    const int khalf = (lane >> 4) * 2;    // K offset for upper half-wave
    const int mbase = (lane >> 4) << 3;   // C/D row offset for upper half-wave

    // Banded gaussian matrix chunks; identical layout serves as horizontal-B
    // (lane&15 = N) and vertical-A (lane&15 = M):  value(k) = w(k - (lane&15))
    v2f Wband[7];
    #pragma unroll
    for (int j = 0; j < 7; ++j) {
        int k = 4 * j + khalf;
        Wband[j].x = wval(k     - nlo, invs);
        Wband[j].y = wval(k + 1 - nlo, invs);
    }

    // ---- horizontal pass: hsum(32x16) = tile(32x28) x Wband(28x16), 2 row-groups
    #pragma unroll
    for (int g = 0; g < 2; ++g) {
        v8f a0 = {}, a1 = {}, a2 = {}, a3 = {}, a4 = {};
        const int r = g * 16 + nlo;
        #pragma unroll
        for (int j = 0; j < 7; ++j) {
            int k = 4 * j + khalf;
            v2f xa = *(const v2f*)&xs[r][k];     // ds_load_b64, aligned
            v2f yb = *(const v2f*)&ys[r][k];
            v2f xx; xx.x = xa.x * xa.x; xx.y = xa.y * xa.y;
            v2f yy; yy.x = yb.x * yb.x; yy.y = yb.y * yb.y;
            v2f xy; xy.x = xa.x * yb.x; xy.y = xa.y * yb.y;
            a0 = WMMA_F32(xa, Wband[j], a0);
            a1 = WMMA_F32(yb, Wband[j], a1);
            a2 = WMMA_F32(xx, Wband[j], a2);
            a3 = WMMA_F32(yy, Wband[j], a3);
            a4 = WMMA_F32(xy, Wband[j], a4);
        }
        // D layout: VGPR v -> row v+mbase, col nlo; store transposed as 2x b128 each
        const int mrow = g * 16 + mbase;
        v8f accs[5] = { a0, a1, a2, a3, a4 };
        #pragma unroll
        for (int q = 0; q < 5; ++q) {
            v4f lo, hi;
            lo.x = accs[q][0]; lo.y = accs[q][1]; lo.z = accs[q][2]; lo.w = accs[q][3];
            hi.x = accs[q][4]; hi.y = accs[q][5]; hi.z = accs[q][6]; hi.w = accs[q][7];
            *(v4f*)&hsT[q][nlo][mrow]     = lo;   // 16B aligned (36*nlo+mrow % 4 == 0)
            *(v4f*)&hsT[q][nlo][mrow + 4] = hi;
        }
    }

    // ---- vertical pass: out(16x16) = Wband^T(16x28) x hsum(28x16), 5 chains
    v8f r0 = {}, r1 = {}, r2 = {}, r3 = {}, r4 = {};
    #pragma unroll
    for (int j = 0; j < 7; ++j) {
        int k = 4 * j + khalf;
        v2f b0 = *(const v2f*)&hsT[0][nlo][k];   // ds_load_b64
        v2f b1 = *(const v2f*)&hsT[1][nlo][k];
        v2f b2 = *(const v2f*)&hsT[2][nlo][k];
        v2f b3 = *(const v2f*)&hsT[3][nlo][k];
        v2f b4 = *(const v2f*)&hsT[4][nlo][k];
        r0 = WMMA_F32(Wband[j], b0, r0);
        r1 = WMMA_F32(Wband[j], b1, r1);
        r2 = WMMA_F32(Wband[j], b2, r2);
        r3 = WMMA_F32(Wband[j], b3, r3);
        r4 = WMMA_F32(Wband[j], b4, r4);
    }

    // ---- per-pixel SSIM / CS, masked accumulate
    const float c1 = c1c2[0], c2 = c1c2[1];
    float ssim_part = 0.f, cs_part = 0.f;
    #pragma unroll
    for (int v = 0; v < 8; ++v) {
        int oy = oy0 + mbase + v;
        int ox = ox0 + nlo;
        float mu1 = r0[v], mu2 = r1[v];
        float mu1s = mu1 * mu1, mu2s = mu2 * mu2, m12 = mu1 * mu2;
        float s11 = r2[v] - mu1s;
        float s22 = r3[v] - mu2s;
        float s12 = r4[v] - m12;
        float cs  = (2.f * s12 + c2) / (s11 + s22 + c2);
        float sv  = (2.f * m12 + c1) / (mu1s + mu2s + c1) * cs;
        bool valid = (oy < mh) && (ox < mw);
        ssim_part += valid ? sv : 0.f;
        cs_part   += valid ? cs : 0.f;
    }
    #pragma unroll
    for (int off = 16; off > 0; off >>= 1) {
        ssim_part += __shfl_down(ssim_part, off, 32);
        cs_part   += __shfl_down(cs_part, off, 32);
    }
    if (lane == 0) {
        atomicAdd(&ssim_acc[level * NB + n], ssim_part);
        atomicAdd(&cs_acc[level * NB + n], cs_part);
    }
}

// ---------------------------------------------------------------------------
// 2x2 average pool, vectorized: b128 loads, b64 stores (all dims even)
// ---------------------------------------------------------------------------
__global__ void pool2_kernel(const float* __restrict__ xi, const float* __restrict__ yi,
                             float* __restrict__ xo, float* __restrict__ yo,
                             int H, int W, long pairs)   // pairs = total_out / 2
{
    const int Ho = H / 2, Wo = W / 2, Wp = Wo / 2;
    for (long i = (long)blockIdx.x * blockDim.x + threadIdx.x; i < pairs;
         i += (long)blockDim.x * gridDim.x) {
        long wp  = i % Wp;
        long rest = i / Wp;
        long h   = rest % Ho;
        long im  = rest / Ho;
        long base = im * (long)H * W + (2 * h) * W + 4 * wp;
        float4 x0 = *(const float4*)&xi[base];
        float4 x1 = *(const float4*)&xi[base + W];
        float4 y0 = *(const float4*)&yi[base];
        float4 y1 = *(const float4*)&yi[base + W];
        float2 ox, oy;
        ox.x = 0.25f * (x0.x + x0.y + x1.x + x1.y);
        ox.y = 0.25f * (x0.z + x0.w + x1.z + x1.w);
        oy.x = 0.25f * (y0.x + y0.y + y1.x + y1.y);
        oy.y = 0.25f * (y0.z + y0.w + y1.z + y1.w);
        *(float2*)&xo[2 * i] = ox;
        *(float2*)&yo[2 * i] = oy;
    }
}

// ---------------------------------------------------------------------------
// min/max reduction of target -> c1, c2  (float4 loads; ntot % 4 == 0)
// ---------------------------------------------------------------------------
__global__ void minmax_partial(const float* __restrict__ t, long nquads, float* __restrict__ part)
{
    __shared__ float smin[256], smax[256];
    float mn = 3.4e38f, mx = -3.4e38f;
    for (long i = (long)blockIdx.x * 256 + threadIdx.x; i < nquads; i += 256L * gridDim.x) {
        float4 v = *(const float4*)&t[4 * i];
        mn = fminf(mn, fminf(fminf(v.x, v.y), fminf(v.z, v.w)));
        mx = fmaxf(mx, fmaxf(fmaxf(v.x, v.y), fmaxf(v.z, v.w)));
    }
    smin[threadIdx.x] = mn; smax[threadIdx.x] = mx;
    __syncthreads();
    for (int sft = 128; sft > 0; sft >>= 1) {
        if ((int)threadIdx.x < sft) {
            smin[threadIdx.x] = fminf(smin[threadIdx.x], smin[threadIdx.x + sft]);
            smax[threadIdx.x] = fmaxf(smax[threadIdx.x], smax[threadIdx.x + sft]);
        }
        __syncthreads();
    }
    if (threadIdx.x == 0) {
        part[2 * blockIdx.x]     = smin[0];
        part[2 * blockIdx.x + 1] = smax[0];
    }
}

__global__ void minmax_final(const float* __restrict__ part, int nb, float* __restrict__ c1c2)
{
    __shared__ float smin[1024], smax[1024];
    float mn = 3.4e38f, mx = -3.4e38f;
    for (int i = threadIdx.x; i < nb; i += blockDim.x) {
        mn = fminf(mn, part[2 * i]);
        mx = fmaxf(mx, part[2 * i + 1]);
    }
    smin[threadIdx.x] = mn; smax[threadIdx.x] = mx;
    __syncthreads();
    for (int sft = (int)blockDim.x / 2; sft > 0; sft >>= 1) {
        if ((int)threadIdx.x < sft) {
            smin[threadIdx.x] = fminf(smin[threadIdx.x], smin[threadIdx.x + sft]);
            smax[threadIdx.x] = fmaxf(smax[threadIdx.x], smax[threadIdx.x + sft]);
        }
        __syncthreads();
    }
    if (threadIdx.x == 0) {
        float mv = smax[0] - smin[0];
        c1c2[0] = (0.01f * mv) * (0.01f * mv);
        c1c2[1] = (0.03f * mv) * (0.03f * mv);
    }
}

__global__ void zero_kernel(float* p, int n)
{
    for (int i = blockIdx.x * blockDim.x + threadIdx.x; i < n; i += blockDim.x * gridDim.x)
        p[i] = 0.f;
}

// ---------------------------------------------------------------------------
// final combine: ms[n] = prod_{l<4} cs_l^{w_l} * ssim^{4*w_4}   (faithful to ref)
// ---------------------------------------------------------------------------
__global__ void finalize_kernel(const float* __restrict__ ssim_acc,
                                const float* __restrict__ cs_acc,
                                float* __restrict__ out)
{
    const int lane = threadIdx.x;   // 32 threads
    const float wts[5] = {0.0448f, 0.2856f, 0.3001f, 0.2363f, 0.1333f};
    float ms;
    {
        float prod = 1.f, ssim_last = 0.f;
        int d = 512;
        #pragma unroll
        for (int l = 0; l < 5; ++l) {
            int m = d - 10;
            float cnt = (float)(NC * m * m);
            float cs = cs_acc[l * NB + lane] / cnt;
            float sv = ssim_acc[l * NB + lane] / cnt;
            if (l < 4) prod *= powf(cs, wts[l]);
            else       ssim_last = sv;
            d >>= 1;
        }
        ms = prod * powf(ssim_last, 4.f * wts[4]);
    }
    #pragma unroll
    for (int off = 16; off > 0; off >>= 1) ms += __shfl_down(ms, off, 32);
    if (lane == 0) out[0] = ms / (float)NB;
}

// ---------------------------------------------------------------------------
extern "C" void kernel_launch(void* const* d_in, const int* in_sizes, int n_in,
                              void* d_out, int out_size, void* d_ws, size_t ws_size,
                              hipStream_t stream)
{
    const float* x0 = (const float*)d_in[0];
    const float* y0 = (const float*)d_in[1];
    const long total0 = (long)in_sizes[1];           // 32*3*512*512

    // workspace layout (floats)
    float* hdr      = (float*)d_ws;                  // [0]=c1 [1]=c2
    float* ssim_acc = hdr + 16;                      // 5*32
    float* cs_acc   = hdr + 256;                     // 5*32
    float* part     = hdr + 512;                     // 2*1024 minmax partials
    float* pyr      = hdr + 4096;

    // pyramid offsets for levels 1..4 (x then y per level)
    long sizes[5];
    sizes[0] = (long)NB * NC * 512 * 512;
    for (int l = 1; l < 5; ++l) sizes[l] = sizes[l - 1] / 4;
    float* xp[5]; float* yp[5];
    long off = 0;
    for (int l = 1; l < 5; ++l) {
        xp[l] = pyr + off; off += sizes[l];
        yp[l] = pyr + off; off += sizes[l];
    }

    // 1) zero accumulators, 2) c1/c2 from target min/max
    zero_kernel<<<1, 256, 0, stream>>>(hdr, 512);
    minmax_partial<<<1024, 256, 0, stream>>>(y0, total0 / 4, part);
    minmax_final<<<1, 1024, 0, stream>>>(part, 1024, hdr);

    // 3) per-level SSIM (WMMA) + pooling
    const float* xc = x0;
    const float* yc = y0;
    int H = 512;
    for (int l = 0; l < 5; ++l) {
        int mh = H - 10;
        dim3 grid((mh + 15) / 16, (mh + 15) / 16, NB * NC);
        ssim_level_kernel<<<grid, 32, 0, stream>>>(xc, yc, H, H, l, hdr, ssim_acc, cs_acc);
        if (l < 5 - 1) {
            long pairs = (long)NB * NC * (H / 2) * (H / 2) / 2;
            int blocks = (int)((pairs + 255) / 256);
            pool2_kernel<<<blocks, 256, 0, stream>>>(xc, yc, xp[l + 1], yp[l + 1], H, H, pairs);
            xc = xp[l + 1];
            yc = yp[l + 1];
            H /= 2;
        }
    }

    // 4) combine to scalar
    finalize_kernel<<<1, 32, 0, stream>>>(ssim_acc, cs_acc, (float*)d_out);
}